// Model_69097433858112
// MI455X (gfx1250) — compile-verified
//
#include <hip/hip_runtime.h>
#include <hip/hip_bf16.h>

typedef __attribute__((ext_vector_type(2))) float v2f;
typedef __attribute__((ext_vector_type(8))) float v8f;
typedef int v4i __attribute__((vector_size(16)));
typedef __attribute__((address_space(1))) v4i* as1_v4i_p;
typedef __attribute__((address_space(3))) v4i* as3_v4i_p;

#define Bb 4
#define Sq 4096
#define Hh 16
#define Pp 64
#define Nn 64
#define Ll 64
#define Cc 64
#define ROWSTR 1024           /* h*p = h*n elements per time step            */
#define LSTR 68               /* padded LDS row stride: 272B, 16B-aligned    */
#define IDX(r, c) ((r) * LSTR + (c))

#if defined(__has_builtin)
#if __has_builtin(__builtin_amdgcn_global_load_async_to_lds_b128)
#define HAS_ASYNC_LDS 1
#endif
#endif

static __device__ __forceinline__ v8f wmma_f32_k4(v2f a, v2f b, v8f c) {
  return __builtin_amdgcn_wmma_f32_16x16x4_f32(false, a, false, b, (short)0, c,
                                               false, false);
}

// Copy one float4 from global to LDS. Async (memory->LDS direct, ASYNCcnt)
// when the toolchain exposes the gfx1250 builtin; VGPR bounce otherwise.
static __device__ __forceinline__ void stage16(const float* g, float* l) {
#ifdef HAS_ASYNC_LDS
  __builtin_amdgcn_global_load_async_to_lds_b128((as1_v4i_p)g, (as3_v4i_p)l, 0,
                                                 0);
#else
  float4 v = *reinterpret_cast<const float4*>(g);
  l[0] = v.x; l[1] = v.y; l[2] = v.z; l[3] = v.w;
#endif
}

static __device__ __forceinline__ void stage_wait() {
#ifdef HAS_ASYNC_LDS
  asm volatile("s_wait_asynccnt 0x0" ::: "memory");
#endif
}

// ---------------------------------------------------------------------------
// Kernel 1: per (b,c,h) chunk — GEMM1 (C·Bᵀ masked by L), GEMM2 (Y_diag),
// GEMM3 (chunk states); fp32 WMMA 16x16x4 throughout.
// ---------------------------------------------------------------------------
__global__ __launch_bounds__(256) void ssd_chunk_kernel(
    const float* __restrict__ X, const float* __restrict__ A,
    const float* __restrict__ Bm, const float* __restrict__ Cm,
    float* __restrict__ Y, float* __restrict__ ws_states,
    float* __restrict__ ws_alast) {
  __shared__ __align__(16) float sX[64 * LSTR];
  __shared__ __align__(16) float sB[64 * LSTR];
  __shared__ __align__(16) float sC[64 * LSTR];
  __shared__ __align__(16) float sM[64 * LSTR];
  __shared__ float sCs[64];
  __shared__ float sDecay[64];

  const int blk = blockIdx.x;
  const int bi = blk / (Cc * Hh);
  const int rem = blk % (Cc * Hh);
  const int ci = rem / Hh;
  const int hi = rem % Hh;
  const int tid = threadIdx.x;
  const int lane = tid & 31;
  const int wave = tid >> 5;

  const size_t base = (((size_t)bi * Sq + (size_t)ci * Ll) * Hh + hi) * Pp;

  // Stage X, B, C chunk tiles (64x64 each): 1024 float4 per matrix.
  for (int idx = tid; idx < 1024; idx += 256) {
    const int row = idx >> 4;
    const int col = (idx & 15) << 2;
    const size_t g = base + (size_t)row * ROWSTR + col;
    stage16(&X[g], &sX[IDX(row, col)]);
    stage16(&Bm[g], &sB[IDX(row, col)]);
    stage16(&Cm[g], &sC[IDX(row, col)]);
  }
  if (tid < 64)
    sCs[tid] = A[((size_t)bi * Sq + (size_t)ci * Ll + tid) * Hh + hi];
  stage_wait();
  __syncthreads();
  if (tid == 0) {  // serial inclusive prefix over 64 values (cheap)
    float s = 0.f;
    for (int i = 0; i < 64; ++i) { s += sCs[i]; sCs[i] = s; }
  }
  __syncthreads();
  if (tid < 64) sDecay[tid] = __expf(sCs[63] - sCs[tid]);
  __syncthreads();

  const int mrow = lane & 15;
  const int koff = (lane >> 4) ? 2 : 0;
  const int rsel = (lane >> 4) ? 8 : 0;

  // ---- GEMM1: G = C · Bᵀ, then M = L ⊙ G into sM (branch-free mask) ----
  for (int t = wave; t < 16; t += 8) {
    const int tr = (t >> 2) << 4, tc = (t & 3) << 4;
    const int am = tr + mrow, bn = tc + mrow;
    v8f acc = {0.f, 0.f, 0.f, 0.f, 0.f, 0.f, 0.f, 0.f};
#pragma unroll
    for (int k = 0; k < 64; k += 4) {
      v2f av, bv;
      av.x = sC[IDX(am, k + koff)];
      av.y = sC[IDX(am, k + koff + 1)];
      bv.x = sB[IDX(bn, k + koff)];          // Bᵀ[k][j] = B[j][k]
      bv.y = sB[IDX(bn, k + koff + 1)];
      acc = wmma_f32_k4(av, bv, acc);
    }
    const int j = tc + mrow;
    const float csj = sCs[j];
#pragma unroll
    for (int r = 0; r < 8; ++r) {
      const int i = tr + r + rsel;
      const float e = __expf(sCs[i] - csj);         // unconditional
      sM[IDX(i, j)] = (j <= i) ? acc[r] * e : 0.f;  // select, no branch
    }
  }
  __syncthreads();

  // ---- GEMM2: Y_diag = M · X, stored directly to global Y ----
  for (int t = wave; t < 16; t += 8) {
    const int tr = (t >> 2) << 4, tc = (t & 3) << 4;
    const int am = tr + mrow, bn = tc + mrow;
    v8f acc = {0.f, 0.f, 0.f, 0.f, 0.f, 0.f, 0.f, 0.f};
#pragma unroll
    for (int k = 0; k < 64; k += 4) {
      v2f av, bv;
      av.x = sM[IDX(am, k + koff)];
      av.y = sM[IDX(am, k + koff + 1)];
      bv.x = sX[IDX(k + koff, bn)];
      bv.y = sX[IDX(k + koff + 1, bn)];
      acc = wmma_f32_k4(av, bv, acc);
    }
#pragma unroll
    for (int r = 0; r < 8; ++r) {
      const int i = tr + r + rsel;
      const int pp = tc + mrow;
      Y[base + (size_t)i * ROWSTR + pp] = acc[r];
    }
  }

  // ---- GEMM3: states[p][n] = Σ_j X[j][p] · (decay[j] · B[j][n]) ----
  float* st =
      ws_states + ((((size_t)bi * Hh + hi) * Cc + ci) << 12);  // p*n = 4096
  for (int t = wave; t < 16; t += 8) {
    const int tr = (t >> 2) << 4, tc = (t & 3) << 4;
    const int am = tr + mrow, bn = tc + mrow;
    v8f acc = {0.f, 0.f, 0.f, 0.f, 0.f, 0.f, 0.f, 0.f};
#pragma unroll
    for (int k = 0; k < 64; k += 4) {
      v2f av, bv;
      av.x = sX[IDX(k + koff, am)];          // Xᵀ[p][j]
      av.y = sX[IDX(k + koff + 1, am)];
      bv.x = sDecay[k + koff] * sB[IDX(k + koff, bn)];
      bv.y = sDecay[k + koff + 1] * sB[IDX(k + koff + 1, bn)];
      acc = wmma_f32_k4(av, bv, acc);
    }
#pragma unroll
    for (int r = 0; r < 8; ++r) {
      const int pp = tr + r + rsel;
      const int nn = tc + mrow;
      st[(pp << 6) + nn] = acc[r];
    }
  }
  if (tid == 0) ws_alast[(bi * Hh + hi) * Cc + ci] = sCs[63];
}

// ---------------------------------------------------------------------------
// Kernel 2: per (b,h) sequential inter-chunk scan, in-place rewrite of the
// workspace: chunk-local states -> entering states S_in[c].
// ---------------------------------------------------------------------------
__global__ __launch_bounds__(256) void ssd_scan_kernel(
    const float* __restrict__ init_states, float* __restrict__ ws_states,
    const float* __restrict__ ws_alast) {
  const int bh = blockIdx.x;  // b*h = 64
  const int tid = threadIdx.x;
  float s[16];
  const float* ip = init_states + ((size_t)bh << 12) + tid * 16;
#pragma unroll
  for (int r = 0; r < 16; ++r) s[r] = ip[r];
  float* basep = ws_states + ((size_t)bh * Cc << 12) + tid * 16;
  const float* al = ws_alast + (size_t)bh * Cc;
  for (int ci = 0; ci < Cc; ++ci) {
    float* p = basep + ((size_t)ci << 12);
    const float dec = __expf(al[ci]);
    float chunk[16];
#pragma unroll
    for (int r = 0; r < 16; ++r) chunk[r] = p[r];   // chunk-local state
#pragma unroll
    for (int r = 0; r < 16; ++r) p[r] = s[r];       // overwrite with S_in
#pragma unroll
    for (int r = 0; r < 16; ++r) s[r] = s[r] * dec + chunk[r];
  }
}

// ---------------------------------------------------------------------------
// Kernel 3: Y += exp(cumsum(A)) ⊙ (C · S_inᵀ)  via fp32 WMMA.
// ---------------------------------------------------------------------------
__global__ __launch_bounds__(256) void ssd_off_kernel(
    const float* __restrict__ A, const float* __restrict__ Cm,
    const float* __restrict__ ws_states, float* __restrict__ Y) {
  __shared__ __align__(16) float sC[64 * LSTR];
  __shared__ __align__(16) float sS[64 * LSTR];  // S_in, [p][n]
  __shared__ float sCs[64];

  const int blk = blockIdx.x;
  const int bi = blk / (Cc * Hh);
  const int rem = blk % (Cc * Hh);
  const int ci = rem / Hh;
  const int hi = rem % Hh;
  const int tid = threadIdx.x;
  const int lane = tid & 31;
  const int wave = tid >> 5;

  const size_t base = (((size_t)bi * Sq + (size_t)ci * Ll) * Hh + hi) * Pp;
  const float* Sin = ws_states + ((((size_t)bi * Hh + hi) * Cc + ci) << 12);

  for (int idx = tid; idx < 1024; idx += 256) {
    const int row = idx >> 4;
    const int col = (idx & 15) << 2;
    stage16(&Cm[base + (size_t)row * ROWSTR + col], &sC[IDX(row, col)]);
    stage16(&Sin[(row << 6) + col], &sS[IDX(row, col)]);
  }
  if (tid < 64)
    sCs[tid] = A[((size_t)bi * Sq + (size_t)ci * Ll + tid) * Hh + hi];
  stage_wait();
  __syncthreads();
  if (tid == 0) {
    float s = 0.f;
    for (int i = 0; i < 64; ++i) { s += sCs[i]; sCs[i] = s; }
  }
  __syncthreads();

  const int mrow = lane & 15;
  const int koff = (lane >> 4) ? 2 : 0;
  const int rsel = (lane >> 4) ? 8 : 0;

  for (int t = wave; t < 16; t += 8) {
    const int tr = (t >> 2) << 4, tc = (t & 3) << 4;
    const int am = tr + mrow, bn = tc + mrow;  // bn = output col = p index
    v8f acc = {0.f, 0.f, 0.f, 0.f, 0.f, 0.f, 0.f, 0.f};
#pragma unroll
    for (int k = 0; k < 64; k += 4) {          // k over state dim n
      v2f av, bv;
      av.x = sC[IDX(am, k + koff)];
      av.y = sC[IDX(am, k + koff + 1)];
      bv.x = sS[IDX(bn, k + koff)];            // S_inᵀ[n][p] = S_in[p][n]
      bv.y = sS[IDX(bn, k + koff + 1)];
      acc = wmma_f32_k4(av, bv, acc);
    }
#pragma unroll
    for (int r = 0; r < 8; ++r) {
      const int i = tr + r + rsel;
      const int pp = tc + mrow;
      const size_t o = base + (size_t)i * ROWSTR + pp;
      Y[o] = Y[o] + __expf(sCs[i]) * acc[r];
    }
  }
}

// ---------------------------------------------------------------------------
extern "C" void kernel_launch(void* const* d_in, const int* in_sizes, int n_in,
                              void* d_out, int out_size, void* d_ws,
                              size_t ws_size, hipStream_t stream) {
  const float* X = (const float*)d_in[0];
  const float* init_states = (const float*)d_in[1];
  const float* A = (const float*)d_in[2];
  const float* Bm = (const float*)d_in[3];
  const float* Cm = (const float*)d_in[4];
  float* Y = (float*)d_out;

  float* ws_states = (float*)d_ws;                       // b*h*c*p*n floats
  float* ws_alast = ws_states + (size_t)Bb * Hh * Cc * Pp * Nn;  // b*h*c

  const int nblk = Bb * Cc * Hh;  // 4096
  ssd_chunk_kernel<<<nblk, 256, 0, stream>>>(X, A, Bm, Cm, Y, ws_states,
                                             ws_alast);
  ssd_scan_kernel<<<Bb * Hh, 256, 0, stream>>>(init_states, ws_states,
                                               ws_alast);
  ssd_off_kernel<<<nblk, 256, 0, stream>>>(A, Cm, ws_states, Y);
}